// ST_Joint_Att33_54073638257087
// MI455X (gfx1250) — compile-verified
//
#include <hip/hip_runtime.h>
#include <hip/hip_bf16.h>
#include <math.h>

// Problem constants (from reference): N=32, C=256, T=512, V=25, CI=64
#define PN 32
#define PC 256
#define PT 512
#define PV 25
#define PCI 64
#define L1 537   // T + V
#define L2C 768  // T + C

typedef float v2f __attribute__((ext_vector_type(2)));
typedef float v8f __attribute__((ext_vector_type(8)));

// ---------------------------------------------------------------------------
// Kernel 1: one pass over x producing
//   cat [n][c][t]      = mean over v            (direct write)
//   cat [n][c][512+v]  = mean over t            (atomic partials over t-tiles)
//   cat2[n][v][t]      = mean over c            (atomic partials over c-chunks)
// Block = (c-chunk of 64) x (t-tile of 64); slab ~410KB stays in L2 so the
// second walk (for the t-mean) re-reads from cache, not HBM.
// ---------------------------------------------------------------------------
#define R_CC 64
#define R_TT 64

__global__ __launch_bounds__(256)
void k_reduce(const float* __restrict__ x, float* __restrict__ cat,
              float* __restrict__ cat2) {
  const int n  = blockIdx.z;
  const int c0 = blockIdx.x * R_CC;
  const int t0 = blockIdx.y * R_TT;
  const int j  = threadIdx.x;
  const int tl = j & 63;   // local t
  const int g  = j >> 6;   // c group 0..3
  const int t  = t0 + tl;

  const float* xn = x + (size_t)n * PC * PT * PV;

  float accC[PV];
#pragma unroll
  for (int v = 0; v < PV; ++v) accC[v] = 0.f;

  // ---- pass 1: row sums (mean over v) + per-thread c-partials ----
  for (int ci = 0; ci < R_CC / 4; ++ci) {
    const int c = c0 + g + 4 * ci;
    const float* row = xn + ((size_t)c * PT + t) * PV;
    float s = 0.f;
#pragma unroll
    for (int v = 0; v < PV; ++v) {
      float xv = row[v];
      s += xv;
      accC[v] += xv;
    }
    cat[((size_t)n * PC + c) * L1 + t] = s * (1.0f / PV);
  }

  // combine the 4 c-groups per t, then atomically accumulate mean-over-C
  __shared__ float lds[256 * PV];  // 25.6 KB
#pragma unroll
  for (int v = 0; v < PV; ++v) lds[j * PV + v] = accC[v];
  __syncthreads();
  if (j < R_TT) {
    for (int v = 0; v < PV; ++v) {
      float s = lds[(0 * 64 + j) * PV + v] + lds[(1 * 64 + j) * PV + v] +
                lds[(2 * 64 + j) * PV + v] + lds[(3 * 64 + j) * PV + v];
      atomicAdd(&cat2[((size_t)n * PV + v) * L2C + (t0 + j)], s * (1.0f / PC));
    }
  }

  // ---- pass 2: mean over t (slab re-read hits L2) ----
  for (int b = j; b < R_CC * PV; b += 256) {
    const int cl = b / PV;
    const int v  = b - PV * cl;
    const int c  = c0 + cl;
    const float* col = xn + ((size_t)c * PT + t0) * PV + v;
    float s = 0.f;
    for (int tt = 0; tt < R_TT; ++tt) s += col[(size_t)tt * PV];
    atomicAdd(&cat[((size_t)n * PC + c) * L1 + PT + v], s * (1.0f / PT));
  }
}

// ---------------------------------------------------------------------------
// Kernel 1b: cat2[n][v][512+c] = cat[n][c][512+v]  (tiny transpose copy)
// ---------------------------------------------------------------------------
__global__ void k_copy_t(const float* __restrict__ cat, float* __restrict__ cat2) {
  int idx = blockIdx.x * blockDim.x + threadIdx.x;  // 32*256*25 = 204800
  if (idx >= PN * PC * PV) return;
  int n = idx / (PC * PV);
  int r = idx - n * (PC * PV);
  int c = r / PV;
  int v = r - c * PV;
  cat2[((size_t)n * PV + v) * L2C + PT + c] =
      cat[((size_t)n * PC + c) * L1 + PT + v];
}

// ---------------------------------------------------------------------------
// Generic batched GEMM via V_WMMA_F32_16X16X4_F32, one wave per 16x16 tile.
//   Out[bat] = epilogue( W(MxK) @ X[bat](KxN) )
// A frag (16x4 f32): lanes 0-15 hold M=0..15; vgpr0 = K {k+2*half},
//                    vgpr1 = K {k+2*half+1}.  B frag mirrors with N on lanes.
//
// Masking note: out-of-range M rows / N cols only pollute D entries we never
// store (WMMA D[m][n] depends only on A row m and B lane-column n), so they
// need only *address* clamping.  Only a K-tail overrun feeds every output and
// requires zero-fill -> compile-time KTAIL flag.
//
// KTAIL=false: K%4==0 and M%16==0 -> branch-free inner loop, A as 8B v2f load.
// EPI 0: BatchNorm + hardswish (params g,b,mean,var)   EPI 1: bias + sigmoid
// ---------------------------------------------------------------------------
template <int EPI, bool KTAIL>
__global__ __launch_bounds__(32)
void k_gemm_wmma(const float* __restrict__ W, const float* __restrict__ X,
                 long xstride, int xoff, int ldx, float* __restrict__ Out,
                 long ostride, int ldo, int M, int K, int Nn,
                 const float* __restrict__ p0, const float* __restrict__ p1,
                 const float* __restrict__ p2, const float* __restrict__ p3) {
  const int lane = threadIdx.x;
  const int l16  = lane & 15;
  const int half = lane >> 4;
  const int m0 = blockIdx.y * 16;
  const int n0 = blockIdx.x * 16;
  const int bat = blockIdx.z;

  const float* Xn = X + (size_t)bat * xstride + xoff;

  const int rowA = m0 + l16;
  const int col  = n0 + l16;
  const int cB   = min(col, Nn - 1);  // address clamp only; garbage cols unstored

  v8f acc = {0.f, 0.f, 0.f, 0.f, 0.f, 0.f, 0.f, 0.f};

  if (!KTAIL) {
    // ---- fast path: no masking at all ----
    const float* Wrow = W + (size_t)rowA * K + 2 * half;
    const float* Xc   = Xn + cB + (size_t)(2 * half) * ldx;
#pragma unroll 4
    for (int k = 0; k < K; k += 4) {
      v2f a = *(const v2f*)(Wrow + k);            // 8B aligned: K%4==0, even off
      float b0 = Xc[(size_t)k * ldx];
      float b1 = Xc[(size_t)k * ldx + ldx];
      v2f b = {b0, b1};
      acc = __builtin_amdgcn_wmma_f32_16x16x4_f32(false, a, false, b, (short)0,
                                                  acc, false, false);
    }
  } else {
    // ---- K-tail path (tiny GEMMs: K=25, M=25) ----
    const int rA = min(rowA, M - 1);
    for (int k = 0; k < K; k += 4) {
      const int ka0 = k + 2 * half;
      const int ka1 = ka0 + 1;
      const int k0c = min(ka0, K - 1);
      const int k1c = min(ka1, K - 1);
      // zero only the K-overrun (the one case that pollutes stored outputs)
      float a0 = W[(size_t)rA * K + k0c];
      float a1 = W[(size_t)rA * K + k1c];
      a0 = (ka0 < K) ? a0 : 0.f;
      a1 = (ka1 < K) ? a1 : 0.f;
      float b0 = Xn[(size_t)k0c * ldx + cB];
      float b1 = Xn[(size_t)k1c * ldx + cB];
      v2f a = {a0, a1};
      v2f b = {b0, b1};
      acc = __builtin_amdgcn_wmma_f32_16x16x4_f32(false, a, false, b, (short)0,
                                                  acc, false, false);
    }
  }

  if (col < Nn) {
#pragma unroll
    for (int e = 0; e < 8; ++e) {
      const int m = m0 + e + 8 * half;
      if (m < M) {
        float y = acc[e];
        if (EPI == 0) {  // BN + hardswish
          float sc = p0[m] * rsqrtf(p3[m] + 1e-5f);
          y = y * sc + (p1[m] - p2[m] * sc);
          float t6 = fminf(fmaxf(y + 3.0f, 0.0f), 6.0f);
          y = y * t6 * (1.0f / 6.0f);
        } else {  // bias + sigmoid
          y = y + p0[m];
          y = 1.0f / (1.0f + __expf(-y));
        }
        Out[(size_t)bat * ostride + (size_t)m * ldo + col] = y;
      }
    }
  }
}

// ---------------------------------------------------------------------------
// Kernel 3: out[n,c,t,v] = 0.5*(att_t[n,c,t]*att_v[n,c,v]
//                              + att_t1[n,v,t]*att_c[n,v,c])
// Streaming 419MB write; att_t1 tile staged in LDS (stride-65 pad, no bank
// conflicts), att_t chunk in LDS for broadcast.
// ---------------------------------------------------------------------------
#define F_TT 64

__global__ __launch_bounds__(256)
void k_combine(const float* __restrict__ att_t, const float* __restrict__ att_v,
               const float* __restrict__ att_t1, const float* __restrict__ att_c,
               float* __restrict__ out) {
  const int n = blockIdx.z;
  const int c = blockIdx.y;
  const int t0 = blockIdx.x * F_TT;
  const int lane = threadIdx.x & 31;
  const int w = threadIdx.x >> 5;

  __shared__ float s1[PV * 65];  // att_t1[n, v, t0:t0+64], padded
  __shared__ float st[F_TT];     // att_t[n, c, t0:t0+64]

  for (int i = threadIdx.x; i < PV * F_TT; i += 256) {
    int v = i >> 6, dt = i & 63;
    s1[v * 65 + dt] = att_t1[((size_t)n * PV + v) * PT + t0 + dt];
  }
  if (threadIdx.x < F_TT)
    st[threadIdx.x] = att_t[((size_t)n * PC + c) * PT + t0 + threadIdx.x];
  __syncthreads();

  const int v = min(lane, PV - 1);
  const float av = att_v[((size_t)n * PC + c) * PV + v];
  const float ac = att_c[((size_t)n * PV + v) * PC + c];
  float* op = out + ((size_t)n * PC + c) * (size_t)PT * PV;

  for (int i = 0; i < F_TT / 8; ++i) {
    const int dt = i * 8 + w;
    const float r = 0.5f * (st[dt] * av + s1[v * 65 + dt] * ac);
    if (lane < PV) op[(size_t)(t0 + dt) * PV + lane] = r;
  }
}

// ---------------------------------------------------------------------------
// Launch
// ---------------------------------------------------------------------------
extern "C" void kernel_launch(void* const* d_in, const int* in_sizes, int n_in,
                              void* d_out, int out_size, void* d_ws,
                              size_t ws_size, hipStream_t stream) {
  (void)in_sizes; (void)n_in; (void)out_size; (void)ws_size;

  const float* x        = (const float*)d_in[0];
  const float* fcn_w    = (const float*)d_in[1];
  const float* bn1_g    = (const float*)d_in[2];
  const float* bn1_b    = (const float*)d_in[3];
  const float* bn1_m    = (const float*)d_in[4];
  const float* bn1_v    = (const float*)d_in[5];
  const float* conv_t_w = (const float*)d_in[6];
  const float* conv_t_b = (const float*)d_in[7];
  const float* conv_v_w = (const float*)d_in[8];
  const float* conv_v_b = (const float*)d_in[9];
  const float* fcn1_w   = (const float*)d_in[10];
  const float* bn2_g    = (const float*)d_in[11];
  const float* bn2_b    = (const float*)d_in[12];
  const float* bn2_m    = (const float*)d_in[13];
  const float* bn2_v    = (const float*)d_in[14];
  const float* conv_t1_w = (const float*)d_in[15];
  const float* conv_t1_b = (const float*)d_in[16];
  const float* conv_v1_w = (const float*)d_in[17];
  const float* conv_v1_b = (const float*)d_in[18];
  float* out = (float*)d_out;

  // workspace layout (floats), ~47 MB total
  float* ws = (float*)d_ws;
  const size_t SZ_CAT  = (size_t)PN * PC * L1;    // 4,399,104
  const size_t SZ_CAT2 = (size_t)PN * PV * L2C;   //   614,400
  const size_t SZ_H    = (size_t)PN * PCI * L1;   // 1,099,776
  const size_t SZ_H2   = (size_t)PN * PV * L2C;   //   614,400
  const size_t SZ_AT   = (size_t)PN * PC * PT;    // 4,194,304
  const size_t SZ_AV   = (size_t)PN * PC * PV;
  const size_t SZ_AT1  = (size_t)PN * PV * PT;

  float* cat    = ws;
  float* cat2   = cat + SZ_CAT;
  float* h      = cat2 + SZ_CAT2;
  float* h2     = h + SZ_H;
  float* att_t  = h2 + SZ_H2;
  float* att_v  = att_t + SZ_AT;
  float* att_t1 = att_v + SZ_AV;
  float* att_c  = att_t1 + SZ_AT1;

  // zero atomic accumulation targets (cat + cat2 contiguous at ws start)
  hipMemsetAsync(ws, 0, (SZ_CAT + SZ_CAT2) * sizeof(float), stream);

  // 1) fused reductions (single HBM pass over x)
  k_reduce<<<dim3(PC / R_CC, PT / R_TT, PN), 256, 0, stream>>>(x, cat, cat2);

  // 1b) transpose t-mean into cat2[:, :, 512:768]
  k_copy_t<<<(PN * PC * PV + 255) / 256, 256, 0, stream>>>(cat, cat2);

  // 2a) h = hardswish(bn1(fcn_w @ cat))           (64 x 537, K=256)
  k_gemm_wmma<0, false><<<dim3(34, PCI / 16, PN), 32, 0, stream>>>(
      fcn_w, cat, (long)PC * L1, 0, L1, h, (long)PCI * L1, L1, PCI, PC, L1,
      bn1_g, bn1_b, bn1_m, bn1_v);

  // 2b) att_t = sigmoid(conv_t_w @ h[:, :512] + b)  (256 x 512, K=64)
  k_gemm_wmma<1, false><<<dim3(PT / 16, PC / 16, PN), 32, 0, stream>>>(
      conv_t_w, h, (long)PCI * L1, 0, L1, att_t, (long)PC * PT, PT, PC, PCI,
      PT, conv_t_b, nullptr, nullptr, nullptr);

  // 2c) att_v = sigmoid(conv_v_w @ h[:, 512:537] + b)  (256 x 25, K=64)
  k_gemm_wmma<1, false><<<dim3(2, PC / 16, PN), 32, 0, stream>>>(
      conv_v_w, h, (long)PCI * L1, PT, L1, att_v, (long)PC * PV, PV, PC, PCI,
      PV, conv_v_b, nullptr, nullptr, nullptr);

  // 2d) h2 = hardswish(bn2(fcn1_w @ cat2))        (25 x 768, K=25)
  k_gemm_wmma<0, true><<<dim3(L2C / 16, 2, PN), 32, 0, stream>>>(
      fcn1_w, cat2, (long)PV * L2C, 0, L2C, h2, (long)PV * L2C, L2C, PV, PV,
      L2C, bn2_g, bn2_b, bn2_m, bn2_v);

  // 2e) att_t1 = sigmoid(conv_t1_w @ h2[:, :512] + b)  (25 x 512, K=25)
  k_gemm_wmma<1, true><<<dim3(PT / 16, 2, PN), 32, 0, stream>>>(
      conv_t1_w, h2, (long)PV * L2C, 0, L2C, att_t1, (long)PV * PT, PT, PV,
      PV, PT, conv_t1_b, nullptr, nullptr, nullptr);

  // 2f) att_c = sigmoid(conv_v1_w @ h2[:, 512:768] + b)  (25 x 256, K=25)
  k_gemm_wmma<1, true><<<dim3(PC / 16, 2, PN), 32, 0, stream>>>(
      conv_v1_w, h2, (long)PV * L2C, PT, L2C, att_c, (long)PV * PC, PC, PV,
      PV, PC, conv_v1_b, nullptr, nullptr, nullptr);

  // 3) final streaming combine (write 419 MB)
  k_combine<<<dim3(PT / F_TT, PC, PN), 256, 0, stream>>>(att_t, att_v, att_t1,
                                                         att_c, out);
}